// DotInteract_40389872451968
// MI455X (gfx1250) — compile-verified
//
#include <hip/hip_runtime.h>

typedef __attribute__((ext_vector_type(2))) float v2f;
typedef __attribute__((ext_vector_type(4))) float v4f;
typedef __attribute__((ext_vector_type(8))) float v8f;
typedef __attribute__((ext_vector_type(4))) int   v4i;

#define NSP 26            // sparse features
#define NN  27            // combined rows
#define DD  128           // feature dim
#define TRI 378           // 27*28/2 upper triangle incl. diagonal
#define OUTW 506          // 128 + 378 floats per output row
#define ROWP 132          // LDS row pitch (floats); 132 % 64 == 4 -> conflict-free ds_load_b64
#define NROWS 28          // 27 data rows + 1 zero row (WMMA N=32 padding)
#define WPB 4             // waves per block
#define LDS_PER_WAVE (NROWS * ROWP)   // 3696 floats = 14784 B

#if defined(__AMDGCN__) && __has_builtin(__builtin_amdgcn_global_load_async_to_lds_b128)
#define HAVE_ASYNC_LDS 1
typedef __attribute__((address_space(1))) v4i ga_v4i;   // global int4
typedef __attribute__((address_space(3))) v4i ls_v4i;   // LDS int4
#else
#define HAVE_ASYNC_LDS 0
#endif

// flat index into the row-major upper triangle (incl. diagonal) of a 27x27 matrix
__device__ __forceinline__ int triu_idx(int r, int c) {
    return r * NN - ((r * (r - 1)) >> 1) + (c - r);
}

__global__ __launch_bounds__(WPB * 32) void dot_interact_kernel(
    const float* __restrict__ dense,
    const float* __restrict__ sparse,
    float* __restrict__ out,
    int nsamples)
{
    __shared__ float smem[WPB * LDS_PER_WAVE];   // 59136 B

    const int lane = threadIdx.x & 31;
    const int wave = threadIdx.x >> 5;
    const long b = (long)blockIdx.x * WPB + wave;
    if (b >= nsamples) return;                   // uniform per wave

    float* wl = smem + wave * LDS_PER_WAVE;      // this wave's slab

    // ---- Stage combined[b] (27 rows x 128 f32) into LDS, row pitch 132 floats ----
    // One async instruction per row: 32 lanes x 16 B = 512 B, perfectly coalesced.
    #pragma unroll
    for (int r = 0; r < NN; ++r) {
        const float* g = (r == 0) ? (dense + b * DD)
                                  : (sparse + (b * NSP + (r - 1)) * (long)DD);
        float* l = wl + r * ROWP + lane * 4;
#if HAVE_ASYNC_LDS
        __builtin_amdgcn_global_load_async_to_lds_b128(
            (ga_v4i*)(g + lane * 4), (ls_v4i*)l, 0, 0);
#else
        *(v4f*)l = *(const v4f*)(g + lane * 4);
#endif
    }
    // Zero row 27 (padding target for WMMA rows 27..31).
    *(v4f*)(wl + 27 * ROWP + lane * 4) = (v4f){0.f, 0.f, 0.f, 0.f};

#if HAVE_ASYNC_LDS
#if __has_builtin(__builtin_amdgcn_s_wait_asynccnt)
    __builtin_amdgcn_s_wait_asynccnt(0);
#else
    asm volatile("s_wait_asynccnt 0x0" ::: "memory");
#endif
#endif
    __syncthreads();

    // ---- Gram matrix via v_wmma_f32_16x16x4_f32 ----
    // Lane mapping (A and B register images identical for a Gram matrix):
    //   lanes 0-15: M/N = lane, K pair {0,1}; lanes 16-31: M/N = lane-16, K pair {2,3}.
    const int half = lane >> 4;
    const int i    = lane & 15;
    const int kb   = half * 2;

    const float* lo = wl + i * ROWP + kb;                          // rows 0..15
    const int rHi   = 16 + i;
    const float* hi = wl + ((rHi < NN) ? rHi : NN) * ROWP + kb;    // rows 16..26, else zero row

    v8f c00 = {};   // G[0:16, 0:16]
    v8f c01 = {};   // G[0:16, 16:32]
    v8f c11 = {};   // G[16:32, 16:32]

    #pragma unroll
    for (int j = 0; j < DD / 4; ++j) {          // 32 steps: 2 ds_load_b64 + 3 WMMA each
        v2f a  = *(const v2f*)(lo + 4 * j);
        v2f bb = *(const v2f*)(hi + 4 * j);
        c00 = __builtin_amdgcn_wmma_f32_16x16x4_f32(false, a,  false, a,  (short)0, c00, false, false);
        c01 = __builtin_amdgcn_wmma_f32_16x16x4_f32(false, a,  false, bb, (short)0, c01, false, false);
        c11 = __builtin_amdgcn_wmma_f32_16x16x4_f32(false, bb, false, bb, (short)0, c11, false, false);
    }

    // ---- Scatter upper triangle into the (now dead) staging slab ----
    // C/D layout: VGPR r, lane -> element (M = r + half*8, N = i).
    #pragma unroll
    for (int r = 0; r < 8; ++r) {
        const int m = r + half * 8;
        if (m <= i)                 wl[triu_idx(m, i)]           = c00[r];  // R=m,    C=i
        if (16 + i < NN)            wl[triu_idx(m, 16 + i)]      = c01[r];  // R=m,    C=16+i
        if (m <= i && 16 + i < NN)  wl[triu_idx(16 + m, 16 + i)] = c11[r];  // R=16+m, C=16+i
    }
    __syncthreads();

    // ---- Output row: [ dense (128) | triu(G) (378) ], pitch 2024 B (8 B aligned) ----
    const float* dsrc = dense + b * DD;
    float* orow = out + b * (long)OUTW;
    ((float2*)orow)[lane]      = ((const float2*)dsrc)[lane];
    ((float2*)orow)[lane + 32] = ((const float2*)dsrc)[lane + 32];

    float* tout = orow + DD;
    #pragma unroll
    for (int t0 = 0; t0 < TRI; t0 += 32) {
        const int t = t0 + lane;
        if (t < TRI) tout[t] = wl[t];           // coalesced b32 stores
    }
}

extern "C" void kernel_launch(void* const* d_in, const int* in_sizes, int n_in,
                              void* d_out, int out_size, void* d_ws, size_t ws_size,
                              hipStream_t stream) {
    const float* dense  = (const float*)d_in[0];   // (B, 128) fp32
    const float* sparse = (const float*)d_in[1];   // (B, 26, 128) fp32
    float* out = (float*)d_out;                    // (B, 506) fp32

    const int nsamples = in_sizes[0] / DD;
    const int blocks = (nsamples + WPB - 1) / WPB;
    dot_interact_kernel<<<blocks, WPB * 32, 0, stream>>>(dense, sparse, out, nsamples);
}